// GraphSCI_58909771432452
// MI455X (gfx1250) — compile-verified
//
#include <hip/hip_runtime.h>
#include <hip/hip_bf16.h>

typedef __attribute__((ext_vector_type(2))) float v2f;
typedef __attribute__((ext_vector_type(8))) float v8f;

// D = A(16x4 f32) * B(4x16 f32) + C(16x16 f32), full-precision f32 WMMA.
__device__ __forceinline__ v8f wmma4(v2f a, v2f b, v8f c) {
    return __builtin_amdgcn_wmma_f32_16x16x4_f32(
        /*neg_a=*/false, a, /*neg_b=*/false, b,
        /*c_mod=*/(short)0, c, /*reuse_a=*/false, /*reuse_b=*/false);
}

// ---------------- phi = relu(X @ W_phi + b_phi) -------------------------
// grid: N/32 blocks, 128 threads (4 waves). Wave w owns output cols [16w,16w+16)
// for TWO 16-row tiles (B fragments reused across both -> 2 WMMA per B load).
__global__ void k_phi(const float* __restrict__ X, const float* __restrict__ Wp,
                      const float* __restrict__ bp, float* __restrict__ phi, int N) {
    __shared__ float xs[32][128];
    const int t = threadIdx.x;
    {   // stage 32x128 X tile (zero-fill OOB rows): 32 floats per thread
        const int row = t >> 2, chunk = t & 3;
        const int grow = blockIdx.x * 32 + row;
        for (int i = 0; i < 8; ++i) {
            const int c = chunk * 32 + i * 4;
            float4 v = make_float4(0.f, 0.f, 0.f, 0.f);
            if (grow < N) v = *(const float4*)(X + (size_t)grow * 128 + c);
            *(float4*)&xs[row][c] = v;
        }
    }
    __syncthreads();

    const int lane = t & 31, wave = t >> 5;
    const int l = lane & 15, hi = lane >> 4;
    const int col = wave * 16 + l;
    v8f accA = {}, accB = {};
    for (int k0 = 0; k0 < 128; k0 += 4) {
        const int ka = k0 + hi * 2;
        v2f aA; aA.x = xs[l][ka];      aA.y = xs[l][ka + 1];
        v2f aB; aB.x = xs[16 + l][ka]; aB.y = xs[16 + l][ka + 1];
        v2f b;  b.x  = Wp[(size_t)ka * 64 + col]; b.y = Wp[(size_t)(ka + 1) * 64 + col];
        accA = wmma4(aA, b, accA);
        accB = wmma4(aB, b, accB);
    }
    const float bias = bp[col];
    for (int r = 0; r < 8; ++r) {
        const int ra = blockIdx.x * 32 + r + hi * 8;
        const int rb = ra + 16;
        if (ra < N) { float v = accA[r] + bias; phi[(size_t)ra * 64 + col] = v > 0.f ? v : 0.f; }
        if (rb < N) { float v = accB[r] + bias; phi[(size_t)rb * 64 + col] = v > 0.f ? v : 0.f; }
    }
}

// ---------------- h = (treat * phi) @ W_gnn -----------------------------
__global__ void k_h(const float* __restrict__ phi, const float* __restrict__ treat,
                    const float* __restrict__ Wg, float* __restrict__ h, int N) {
    __shared__ float ps[32][64];
    const int t = threadIdx.x;
    {   // stage 32x64 (treat*phi) tile: 16 floats per thread
        const int row = t >> 2, chunk = t & 3;
        const int grow = blockIdx.x * 32 + row;
        const float tt = (grow < N) ? treat[grow] : 0.f;
        for (int i = 0; i < 4; ++i) {
            const int c = chunk * 16 + i * 4;
            float4 v = make_float4(0.f, 0.f, 0.f, 0.f);
            if (grow < N) v = *(const float4*)(phi + (size_t)grow * 64 + c);
            v.x *= tt; v.y *= tt; v.z *= tt; v.w *= tt;
            *(float4*)&ps[row][c] = v;
        }
    }
    __syncthreads();

    const int lane = t & 31, wave = t >> 5;
    const int l = lane & 15, hi = lane >> 4;
    const int col = wave * 16 + l;
    v8f accA = {}, accB = {};
    for (int k0 = 0; k0 < 64; k0 += 4) {
        const int ka = k0 + hi * 2;
        v2f aA; aA.x = ps[l][ka];      aA.y = ps[l][ka + 1];
        v2f aB; aB.x = ps[16 + l][ka]; aB.y = ps[16 + l][ka + 1];
        v2f b;  b.x  = Wg[(size_t)ka * 64 + col]; b.y = Wg[(size_t)(ka + 1) * 64 + col];
        accA = wmma4(aA, b, accA);
        accB = wmma4(aB, b, accB);
    }
    for (int r = 0; r < 8; ++r) {
        const int ra = blockIdx.x * 32 + r + hi * 8;
        const int rb = ra + 16;
        if (ra < N) h[(size_t)ra * 64 + col] = accA[r];
        if (rb < N) h[(size_t)rb * 64 + col] = accB[r];
    }
}

// ---------------- GCN degree / aggregation ------------------------------
__global__ void k_init(float* __restrict__ agg, float* __restrict__ deg, int N) {
    const size_t i = (size_t)blockIdx.x * blockDim.x + threadIdx.x;
    if (i < (size_t)N * 64) agg[i] = 0.f;
    if (i < (size_t)N) deg[i] = 1.0f;   // self-loop
}

__global__ void k_degacc(const int* __restrict__ dst, float* __restrict__ deg, int E) {
    const int e = blockIdx.x * blockDim.x + threadIdx.x;
    if (e < E)
        __hip_atomic_fetch_add(&deg[dst[e]], 1.0f, __ATOMIC_RELAXED, __HIP_MEMORY_SCOPE_AGENT);
}

__global__ void k_dinv(float* __restrict__ deg, int N) {
    const int i = blockIdx.x * blockDim.x + threadIdx.x;
    if (i < N) deg[i] = rsqrtf(deg[i]);   // deg >= 1 always (self-loop)
}

// One wave32 per edge: lane handles 2 of the 64 message floats.
__global__ void k_scatter(const int* __restrict__ src, const int* __restrict__ dst,
                          const float* __restrict__ dinv, const float* __restrict__ h,
                          float* __restrict__ agg, int E) {
    const int lane = threadIdx.x & 31;
    const int wave = threadIdx.x >> 5;
    const int e = blockIdx.x * 8 + wave;
    if (e >= E) return;
    const int s = src[e], d = dst[e];
    const float w = dinv[s] * dinv[d];
    const float2 hv = *(const float2*)(h + (size_t)s * 64 + lane * 2);
    float* ap = agg + (size_t)d * 64 + lane * 2;
    __hip_atomic_fetch_add(ap + 0, w * hv.x, __ATOMIC_RELAXED, __HIP_MEMORY_SCOPE_AGENT);
    __hip_atomic_fetch_add(ap + 1, w * hv.y, __ATOMIC_RELAXED, __HIP_MEMORY_SCOPE_AGENT);
}

// ---------------- output head -------------------------------------------
// rep = [phi | agg + dinv^2*h + b_gnn]; y00 = relu(rep@W00 + b00) (i=1 slice),
// y10 likewise; y0 = y00@W01+b01, y1 = y10@W11+b11.
// grid: N/32 blocks, 256 threads (8 waves x 16-col tiles). Each wave keeps
// 4 accumulators (2 row tiles x 2 weight matrices) so each B fragment feeds
// 2 WMMAs -> 4 WMMA per 4 global b32 loads.
__global__ void k_out(const float* __restrict__ phi, const float* __restrict__ h,
                      const float* __restrict__ agg, const float* __restrict__ dinv,
                      const float* __restrict__ W00, const float* __restrict__ b00,
                      const float* __restrict__ W10, const float* __restrict__ b10,
                      const float* __restrict__ W01, const float* __restrict__ b01,
                      const float* __restrict__ W11, const float* __restrict__ b11,
                      const float* __restrict__ bg,
                      float* __restrict__ y1, float* __restrict__ y0, int N) {
    __shared__ float rep[32][128];
    __shared__ float y0s[32][128];
    __shared__ float y1s[32][128];
    const int t = threadIdx.x;
    {   // stage rep_post tile: 16 elements per thread
        const int row = t >> 3, seg = t & 7;
        const int grow = blockIdx.x * 32 + row;
        const bool valid = grow < N;
        const float di = valid ? dinv[grow] : 0.f;
        const float dself = di * di;
        for (int i = 0; i < 16; ++i) {
            const int c = seg * 16 + i;
            float v = 0.f;
            if (valid) {
                if (c < 64) v = phi[(size_t)grow * 64 + c];
                else {
                    const int cc = c - 64;
                    v = agg[(size_t)grow * 64 + cc] + dself * h[(size_t)grow * 64 + cc] + bg[cc];
                }
            }
            rep[row][c] = v;
        }
    }
    __syncthreads();

    const int lane = t & 31, wave = t >> 5;
    const int l = lane & 15, hi = lane >> 4;
    const int col = wave * 16 + l;
    v8f acc0A = {}, acc0B = {}, acc1A = {}, acc1B = {};
    for (int k0 = 0; k0 < 128; k0 += 4) {
        const int ka = k0 + hi * 2;
        v2f aA; aA.x = rep[l][ka];      aA.y = rep[l][ka + 1];
        v2f aB; aB.x = rep[16 + l][ka]; aB.y = rep[16 + l][ka + 1];
        v2f b0; b0.x = W00[(size_t)ka * 128 + col]; b0.y = W00[(size_t)(ka + 1) * 128 + col];
        v2f b1; b1.x = W10[(size_t)ka * 128 + col]; b1.y = W10[(size_t)(ka + 1) * 128 + col];
        acc0A = wmma4(aA, b0, acc0A);
        acc0B = wmma4(aB, b0, acc0B);
        acc1A = wmma4(aA, b1, acc1A);
        acc1B = wmma4(aB, b1, acc1B);
    }
    const float bb0 = b00[col], bb1 = b10[col];
    for (int r = 0; r < 8; ++r) {
        const int rrA = r + hi * 8;
        const int rrB = rrA + 16;
        float v;
        v = acc0A[r] + bb0; y0s[rrA][col] = v > 0.f ? v : 0.f;
        v = acc0B[r] + bb0; y0s[rrB][col] = v > 0.f ? v : 0.f;
        v = acc1A[r] + bb1; y1s[rrA][col] = v > 0.f ? v : 0.f;
        v = acc1B[r] + bb1; y1s[rrB][col] = v > 0.f ? v : 0.f;
    }
    __syncthreads();

    if (t < 32) {                       // y0_pred rows
        const int grow = blockIdx.x * 32 + t;
        if (grow < N) {
            float s = b01[0];
            for (int c = 0; c < 128; ++c) s += y0s[t][c] * W01[c];
            y0[grow] = s;
        }
    } else if (t < 64) {                // y1_pred rows
        const int rr = t - 32;
        const int grow = blockIdx.x * 32 + rr;
        if (grow < N) {
            float s = b11[0];
            for (int c = 0; c < 128; ++c) s += y1s[rr][c] * W11[c];
            y1[grow] = s;
        }
    }
}

extern "C" void kernel_launch(void* const* d_in, const int* in_sizes, int n_in,
                              void* d_out, int out_size, void* d_ws, size_t ws_size,
                              hipStream_t stream) {
    const float* X     = (const float*)d_in[0];
    const float* treat = (const float*)d_in[1];
    const int*   edge  = (const int*)d_in[2];      // [2,E] int32
    const float* Wp    = (const float*)d_in[3];
    const float* bp    = (const float*)d_in[4];
    const float* Wg    = (const float*)d_in[5];
    const float* bg    = (const float*)d_in[6];
    const float* W00   = (const float*)d_in[7]  + 128 * 128;  // i=1 slice
    const float* b00   = (const float*)d_in[8]  + 128;
    const float* W10   = (const float*)d_in[9]  + 128 * 128;
    const float* b10   = (const float*)d_in[10] + 128;
    const float* W01   = (const float*)d_in[11];
    const float* b01   = (const float*)d_in[12];
    const float* W11   = (const float*)d_in[13];
    const float* b11   = (const float*)d_in[14];

    const int N = in_sizes[1];
    const int E = in_sizes[2] / 2;

    float* out = (float*)d_out;
    float* y1  = out;               // return order: (y1_pred, y0_pred, phi_x)
    float* y0  = out + N;
    float* phi = out + 2 * (size_t)N;

    float* ws  = (float*)d_ws;
    float* h   = ws;                        // N*64
    float* agg = ws + (size_t)N * 64;       // N*64
    float* deg = ws + (size_t)N * 128;      // N  (becomes dinv)

    const int nb32 = (N + 31) / 32;

    k_phi<<<nb32, 128, 0, stream>>>(X, Wp, bp, phi, N);
    k_h<<<nb32, 128, 0, stream>>>(phi, treat, Wg, h, N);
    k_init<<<(int)(((size_t)N * 64 + 255) / 256), 256, 0, stream>>>(agg, deg, N);
    k_degacc<<<(E + 255) / 256, 256, 0, stream>>>(edge + E, deg, E);
    k_dinv<<<(N + 255) / 256, 256, 0, stream>>>(deg, N);
    k_scatter<<<(E + 7) / 8, 256, 0, stream>>>(edge, edge + E, deg, h, agg, E);
    k_out<<<nb32, 256, 0, stream>>>(phi, h, agg, deg, W00, b00, W10, b10,
                                    W01, b01, W11, b11, bg, y1, y0, N);
}